// GAT_33217277067911
// MI455X (gfx1250) — compile-verified
//
#include <hip/hip_runtime.h>
#include <hip/hip_bf16.h>

#define NN 50000
#define NE 800000
#define ET (NE + NN)   // edges + self loops

typedef __attribute__((ext_vector_type(2))) float v2f;
typedef __attribute__((ext_vector_type(8))) float v8f;
typedef __attribute__((__vector_size__(4 * sizeof(int)))) int b128_t;

#if defined(__has_builtin)
#  if __has_builtin(__builtin_amdgcn_global_load_async_to_lds_b128)
#    define GAT_ASYNC_LDS 1
#  endif
#endif
#ifndef GAT_ASYNC_LDS
#  define GAT_ASYNC_LDS 0
#endif

// ---------------------------------------------------------------------------
// GEMM: Out[N,M] = X[N,K] * W[K,M], M == 128, K in {64,128}, N % 16 == 0.
// One block (8 waves) per 16-row tile; wave w computes the 16x16 tile at
// columns [16w,16w+16). X tile staged in LDS (async global->LDS when
// available); K-loop steps of 4 with v_wmma_f32_16x16x4_f32.
// A frag (16x4 f32): lanes 0-15 hold M=lane, v0=K+0,v1=K+1; lanes 16-31 K+2/K+3.
// B frag (4x16 f32): mirrored — v0 = W[k+ksel][n], v1 = W[k+ksel+1][n].
// C/D: VGPR i -> row i (lanes 0-15) / row 8+i (lanes 16-31), col = lane&15.
// ---------------------------------------------------------------------------
__global__ __launch_bounds__(256) void gat_gemm_wmma(
    const float* __restrict__ X, const float* __restrict__ W,
    float* __restrict__ Out, int N, int K, int M)
{
    __shared__ float sX[16 * 128];
    const int rowBase = blockIdx.x * 16;
    const int tid = threadIdx.x;

    // stage the 16xK X tile into LDS at 128-bit granularity (rows fully valid:
    // N is a multiple of 16 for this workload)
    const int Kv = K >> 2;                      // float4s per row (16 or 32)
    for (int i = tid; i < 16 * Kv; i += 256) {
        int r = i / Kv;
        int c = i - r * Kv;
        const float* gp = X + (size_t)(rowBase + r) * K + c * 4;
#if GAT_ASYNC_LDS
        __builtin_amdgcn_global_load_async_to_lds_b128(
            (__attribute__((address_space(1))) b128_t*)(__UINTPTR_TYPE__)gp,
            (__attribute__((address_space(3))) b128_t*)&sX[r * K + c * 4],
            0, 0);
#else
        *(float4*)&sX[r * K + c * 4] = *(const float4*)gp;
#endif
    }
#if GAT_ASYNC_LDS
    asm volatile("s_wait_asynccnt 0x0" ::: "memory");
#endif
    __syncthreads();

    const int lane = tid & 31;
    const int wave = tid >> 5;          // 8 waves -> 8 column tiles (M == 128)
    const int m    = lane & 15;
    const int ksel = (lane >> 4) * 2;
    const int col  = wave * 16 + m;

    v8f acc = {};
    for (int k = 0; k < K; k += 4) {
        v2f a, b;
        a.x = sX[m * K + k + ksel];
        a.y = sX[m * K + k + ksel + 1];
        b.x = W[(size_t)(k + ksel) * M + col];
        b.y = W[(size_t)(k + ksel + 1) * M + col];
        acc = __builtin_amdgcn_wmma_f32_16x16x4_f32(
            false, a, false, b, (short)0, acc, false, false);
    }

    const int rowOff = (lane >> 4) * 8;
#pragma unroll
    for (int i = 0; i < 8; ++i) {
        int row = rowBase + rowOff + i;
        if (row < N) Out[(size_t)row * M + col] = acc[i];
    }
}

// ---------------------------------------------------------------------------
// alpha_s[n,h] = <H[n,h,:], a_s[h,:]>, likewise alpha_d.  H = 2, C = 64.
// ---------------------------------------------------------------------------
__global__ __launch_bounds__(256) void gat_alpha(
    const float* __restrict__ Hf, const float* __restrict__ avs,
    const float* __restrict__ avd, float* __restrict__ asOut,
    float* __restrict__ adOut, int N)
{
    int idx = blockIdx.x * blockDim.x + threadIdx.x;
    if (idx >= N * 2) return;
    int n = idx >> 1, h = idx & 1;
    const float* hp  = Hf + (size_t)n * 128 + h * 64;
    const float* vs  = avs + h * 64;
    const float* vd  = avd + h * 64;
    float ss = 0.f, sd = 0.f;
#pragma unroll 8
    for (int c = 0; c < 64; ++c) {
        float v = hp[c];
        ss += v * vs[c];
        sd += v * vd[c];
    }
    asOut[idx] = ss;
    adOut[idx] = sd;
}

__global__ __launch_bounds__(256) void gat_fill(float* __restrict__ p, float v, int n)
{
    int i = blockIdx.x * blockDim.x + threadIdx.x;
    if (i < n) p[i] = v;
}

__device__ __forceinline__ void edge_sd(const long long* __restrict__ s,
                                        const long long* __restrict__ d,
                                        int e, int& si, int& di)
{
    if (e < NE) { si = (int)s[e]; di = (int)d[e]; }
    else        { si = e - NE;    di = e - NE;    }
}

__device__ __forceinline__ float leaky02(float x)
{
    return x > 0.f ? x : 0.2f * x;
}

// sign-aware float atomic max via integer atomics (init must be -inf)
__device__ __forceinline__ void atomicMaxF32(float* addr, float val)
{
    if (val >= 0.f) atomicMax((int*)addr, __float_as_int(val));
    else            atomicMin((unsigned int*)addr, (unsigned int)__float_as_int(val));
}

// ---------------------------------------------------------------------------
// segment max of leaky-relu logits grouped by dst
// ---------------------------------------------------------------------------
__global__ __launch_bounds__(256) void gat_edge_max(
    const long long* __restrict__ src, const long long* __restrict__ dst,
    const float* __restrict__ aS, const float* __restrict__ aD,
    float* __restrict__ mBuf)
{
    int e = blockIdx.x * blockDim.x + threadIdx.x;
    if (e >= ET) return;
    int si, di;
    edge_sd(src, dst, e, si, di);
#pragma unroll
    for (int h = 0; h < 2; ++h) {
        float l = leaky02(aS[si * 2 + h] + aD[di * 2 + h]);
        atomicMaxF32(&mBuf[di * 2 + h], l);
    }
}

// ---------------------------------------------------------------------------
// e = exp(logit - m[dst]); stash per-edge e and segment-sum into s[dst]
// ---------------------------------------------------------------------------
__global__ __launch_bounds__(256) void gat_edge_exp(
    const long long* __restrict__ src, const long long* __restrict__ dst,
    const float* __restrict__ aS, const float* __restrict__ aD,
    const float* __restrict__ mBuf, float* __restrict__ eBuf,
    float* __restrict__ sBuf)
{
    int e = blockIdx.x * blockDim.x + threadIdx.x;
    if (e >= ET) return;
    int si, di;
    edge_sd(src, dst, e, si, di);
#pragma unroll
    for (int h = 0; h < 2; ++h) {
        float l  = leaky02(aS[si * 2 + h] + aD[di * 2 + h]);
        float ex = expf(l - mBuf[di * 2 + h]);
        eBuf[(size_t)e * 2 + h] = ex;
        atomicAdd(&sBuf[di * 2 + h], ex);
    }
}

// ---------------------------------------------------------------------------
// out[dst,h,c] += (e/s[dst,h]) * H[src,h,c].  One wave per edge:
// lanes 0-15 -> head 0, lanes 16-31 -> head 1, float4 per lane (4 channels).
// ---------------------------------------------------------------------------
__global__ __launch_bounds__(256) void gat_aggregate(
    const long long* __restrict__ src, const long long* __restrict__ dst,
    const float* __restrict__ Hf, const float* __restrict__ eBuf,
    const float* __restrict__ sBuf, float* __restrict__ accum)
{
    int gid  = blockIdx.x * blockDim.x + threadIdx.x;
    int e    = gid >> 5;
    int lane = gid & 31;
    if (e >= ET) return;
    int si, di;
    edge_sd(src, dst, e, si, di);
    int h = lane >> 4;
    float coef = eBuf[(size_t)e * 2 + h] / sBuf[di * 2 + h];
    int ch = h * 64 + (lane & 15) * 4;          // channel in [0,128)
    const float4 hv = *(const float4*)&Hf[(size_t)si * 128 + ch];
    float* o = &accum[(size_t)di * 128 + ch];
    atomicAdd(o + 0, coef * hv.x);
    atomicAdd(o + 1, coef * hv.y);
    atomicAdd(o + 2, coef * hv.z);
    atomicAdd(o + 3, coef * hv.w);
}

// out1 = accum + b1 (concat layout, width 128), in place
__global__ __launch_bounds__(256) void gat_bias128(
    float* __restrict__ P, const float* __restrict__ b, int N)
{
    int i = blockIdx.x * blockDim.x + threadIdx.x;
    if (i >= N * 128) return;
    P[i] += b[i & 127];
}

// final: mean over 2 heads + b2  -> [N,64]
__global__ __launch_bounds__(256) void gat_finalize(
    const float* __restrict__ Q, const float* __restrict__ b,
    float* __restrict__ out, int N)
{
    int i = blockIdx.x * blockDim.x + threadIdx.x;
    if (i >= N * 64) return;
    int n = i >> 6, c = i & 63;
    const float* q = Q + (size_t)n * 128;
    out[i] = 0.5f * (q[c] + q[64 + c]) + b[c];
}

extern "C" void kernel_launch(void* const* d_in, const int* in_sizes, int n_in,
                              void* d_out, int out_size, void* d_ws, size_t ws_size,
                              hipStream_t stream)
{
    const float*     x    = (const float*)d_in[0];
    const long long* edge = (const long long*)d_in[1];  // int64 [2, NE]
    const float*     W1   = (const float*)d_in[2];
    const float*     as1  = (const float*)d_in[3];
    const float*     ad1  = (const float*)d_in[4];
    const float*     b1   = (const float*)d_in[5];
    const float*     W2   = (const float*)d_in[6];
    const float*     as2  = (const float*)d_in[7];
    const float*     ad2  = (const float*)d_in[8];
    const float*     b2   = (const float*)d_in[9];
    float*           out  = (float*)d_out;

    const long long* src = edge;
    const long long* dst = edge + NE;

    // workspace carve-up (floats)
    float* ws   = (float*)d_ws;
    float* Hbuf = ws;                                  // [NN,128] projected feats
    float* Pbuf = Hbuf + (size_t)NN * 128;             // [NN,128] layer1 out / layer2 in
    float* Qbuf = Pbuf + (size_t)NN * 128;             // [NN,128] layer2 accum
    float* aS   = Qbuf + (size_t)NN * 128;             // [NN,2]
    float* aD   = aS + (size_t)NN * 2;                 // [NN,2]
    float* mB   = aD + (size_t)NN * 2;                 // [NN,2]
    float* sB   = mB + (size_t)NN * 2;                 // [NN,2]
    float* eB   = sB + (size_t)NN * 2;                 // [ET,2]

    const int gGemm = (NN + 15) / 16;
    const int gN2   = (NN * 2 + 255) / 256;
    const int gN128 = (NN * 128 + 255) / 256;
    const int gN64  = (NN * 64 + 255) / 256;
    const int gE    = (ET + 255) / 256;
    const int gEagg = (ET * 32 + 255) / 256;

    // ---------------- layer 1 ----------------
    gat_gemm_wmma<<<gGemm, 256, 0, stream>>>(x, W1, Hbuf, NN, 64, 128);
    gat_alpha<<<gN2, 256, 0, stream>>>(Hbuf, as1, ad1, aS, aD, NN);
    gat_fill<<<gN2, 256, 0, stream>>>(mB, -INFINITY, NN * 2);
    gat_fill<<<gN2, 256, 0, stream>>>(sB, 0.0f, NN * 2);
    gat_fill<<<gN128, 256, 0, stream>>>(Pbuf, 0.0f, NN * 128);
    gat_edge_max<<<gE, 256, 0, stream>>>(src, dst, aS, aD, mB);
    gat_edge_exp<<<gE, 256, 0, stream>>>(src, dst, aS, aD, mB, eB, sB);
    gat_aggregate<<<gEagg, 256, 0, stream>>>(src, dst, Hbuf, eB, sB, Pbuf);
    gat_bias128<<<gN128, 256, 0, stream>>>(Pbuf, b1, NN);

    // ---------------- layer 2 ----------------
    gat_gemm_wmma<<<gGemm, 256, 0, stream>>>(Pbuf, W2, Hbuf, NN, 128, 128);
    gat_alpha<<<gN2, 256, 0, stream>>>(Hbuf, as2, ad2, aS, aD, NN);
    gat_fill<<<gN2, 256, 0, stream>>>(mB, -INFINITY, NN * 2);
    gat_fill<<<gN2, 256, 0, stream>>>(sB, 0.0f, NN * 2);
    gat_fill<<<gN128, 256, 0, stream>>>(Qbuf, 0.0f, NN * 128);
    gat_edge_max<<<gE, 256, 0, stream>>>(src, dst, aS, aD, mB);
    gat_edge_exp<<<gE, 256, 0, stream>>>(src, dst, aS, aD, mB, eB, sB);
    gat_aggregate<<<gEagg, 256, 0, stream>>>(src, dst, Hbuf, eB, sB, Qbuf);
    gat_finalize<<<gN64, 256, 0, stream>>>(Qbuf, b2, out, NN);

    (void)in_sizes; (void)n_in; (void)out_size; (void)ws_size;
}